// MultiResolutionHashEncoding2D_78975858639608
// MI455X (gfx1250) — compile-verified
//
#include <hip/hip_runtime.h>
#include <math.h>
#include <stdint.h>

// ---------------------------------------------------------------------------
// MultiResolutionHashEncoding2D for gfx1250 (MI455X)
// Memory-bound gather kernel: no matmul content -> no WMMA possible.
// CDNA5-specific paths used:
//   * global_load_async_to_lds_b64 (+ s_wait_asynccnt) to stage coords in LDS
//   * non-temporal streaming stores (th:TH_STORE_NT) for the 128MB output so
//     the 64MB hash table stays resident in the 192MB L2
// ---------------------------------------------------------------------------

#define TABLE_SIZE   524288          // 2^19 -> modulo is a mask
#define TABLE_MASK   524287u
#define NUM_LEVELS   16
#define HASH_PRIME   2654435761u
#define BLOCK        512
#define PTS_PER_BLK  (BLOCK / NUM_LEVELS)   // 32 points per block

typedef float v2f __attribute__((ext_vector_type(2)));

struct ScaleArr { float s[NUM_LEVELS]; };

__global__ __launch_bounds__(BLOCK)
void MultiResolutionHashEncoding2D_kernel(const v2f* __restrict__ x,
                                          const v2f* __restrict__ table,
                                          v2f* __restrict__ out,
                                          ScaleArr sc, int n_points)
{
    __shared__ v2f s_x[PTS_PER_BLK];

    const int tid       = threadIdx.x;
    const int block_pt0 = blockIdx.x * PTS_PER_BLK;

    // --- Stage 32 points (256B) into LDS with the CDNA5 async copy engine.
    // Wave 0 issues one async b64 per lane; ASYNCcnt tracks completion.
    if (tid < 32) {
        unsigned long long gaddr =
            (unsigned long long)(uintptr_t)(x + block_pt0 + tid);
        // Generic pointers to LDS carry the workgroup LDS offset in addr[31:0]
        unsigned ldsoff = (unsigned)(uintptr_t)(&s_x[tid]);
        if (block_pt0 + tid < n_points) {
            asm volatile("global_load_async_to_lds_b64 %0, %1, off"
                         :: "v"(ldsoff), "v"(gaddr) : "memory");
        }
        asm volatile("s_wait_asynccnt 0" ::: "memory");
    }
    __syncthreads();

    const int lvl      = tid & (NUM_LEVELS - 1);
    const int pt_local = tid >> 4;
    const int pt       = block_pt0 + pt_local;
    if (pt >= n_points) return;

    const v2f   xy    = s_x[pt_local];        // broadcast LDS read
    const float scale = sc.s[lvl];

    const float sx = xy.x * scale;
    const float sy = xy.y * scale;
    const float fx = floorf(sx), fy = floorf(sy);
    const float cx = ceilf(sx),  cy = ceilf(sy);

    const unsigned ufx = (unsigned)(int)fx, ufy = (unsigned)(int)fy;
    const unsigned ucx = (unsigned)(int)cx, ucy = (unsigned)(int)cy;

    // h = vx ^ (vy * PRIME); TABLE_SIZE is 2^19 so % -> mask
    const unsigned hcy  = ucy * HASH_PRIME;
    const unsigned hfy  = ufy * HASH_PRIME;
    const unsigned base = (unsigned)lvl << 19;   // lvl * TABLE_SIZE
    const unsigned i0 = base + ((ucx ^ hcy) & TABLE_MASK);  // (cx, cy)
    const unsigned i1 = base + ((ucx ^ hfy) & TABLE_MASK);  // (cx, fy)
    const unsigned i2 = base + ((ufx ^ hcy) & TABLE_MASK);  // (fx, cy)
    const unsigned i3 = base + ((ufx ^ hfy) & TABLE_MASK);  // (fx, fy)

    // Regular-temporal gathers: 64MB table is L2-resident on MI455X (192MB L2)
    const v2f f0 = table[i0];
    const v2f f1 = table[i1];
    const v2f f2 = table[i2];
    const v2f f3 = table[i3];

    const float ox = sx - fx, oy = sy - fy;
    const float mx = 1.0f - ox, my = 1.0f - oy;

    v2f r;
    r.x = (f0.x * ox + f3.x * mx) * oy + (f1.x * ox + f2.x * mx) * my;
    r.y = (f0.y * ox + f3.y * mx) * oy + (f1.y * ox + f2.y * mx) * my;

    // Streaming output (written once, never re-read): NT store keeps the hash
    // table hot in L2. Index = pt*16 + lvl = blockIdx.x*512 + tid (coalesced).
    __builtin_nontemporal_store(r, &out[(size_t)pt * NUM_LEVELS + lvl]);
}

extern "C" void kernel_launch(void* const* d_in, const int* in_sizes, int n_in,
                              void* d_out, int out_size, void* d_ws, size_t ws_size,
                              hipStream_t stream)
{
    const v2f* x     = (const v2f*)d_in[0];
    const v2f* table = (const v2f*)d_in[1];
    v2f*       out   = (v2f*)d_out;
    const int n_points = in_sizes[0] / 2;

    // scalings[l] = floor(16 * growth^l), growth = exp((ln1024 - ln16)/15)
    ScaleArr sc;
    const double growth = exp((log(1024.0) - log(16.0)) / 15.0);
    for (int l = 0; l < NUM_LEVELS; ++l)
        sc.s[l] = floorf(16.0f * powf((float)growth, (float)l));

    const int blocks = (n_points + PTS_PER_BLK - 1) / PTS_PER_BLK;
    hipLaunchKernelGGL(MultiResolutionHashEncoding2D_kernel,
                       dim3(blocks), dim3(BLOCK), 0, stream,
                       x, table, out, sc, n_points);
}